// NMS_50216757624982
// MI455X (gfx1250) — compile-verified
//
#include <hip/hip_runtime.h>
#include <stdint.h>

// Problem geometry
#define BATCH 16
#define CHAN  80
#define Hh    128
#define Ww    128
#define NPLANE (BATCH*CHAN)          // 1280 planes
#define PLANE_PX (Hh*Ww)             // 16384 px = 64 KB fp32
#define MAX_DET 16384
#define NEGINF (-__builtin_huge_valf())

typedef uint32_t u32x4 __attribute__((ext_vector_type(4)));
typedef uint32_t u32x8 __attribute__((ext_vector_type(8)));

// ---------------------------------------------------------------------------
// LDS byte offset of a __shared__ object: low 32 bits of the generic pointer
// (LDS aperture: addr[31:0] is the LDS offset, ISA §10.2).
__device__ __forceinline__ uint32_t lds_off(const void* p) {
    return (uint32_t)(uintptr_t)p;
}

// ---------------------------------------------------------------------------
// Kernel 0: fill output with -1 (reference pads nonzero() with fill_value=-1)
__global__ void fill_neg1(int* __restrict__ out, int n) {
    int i = blockIdx.x * 256 + threadIdx.x;
    if (i < n) out[i] = -1;
}

// ---------------------------------------------------------------------------
// Span-max steps over one plane held in LDS.  End-clamped forward spans:
// fN[i] = max over [i, min(i+N-1, 127)], built by shifted-max doubling.
__device__ __forceinline__ void vstep(const float* in, float* out, int s, int tid) {
#pragma unroll 8
    for (int i = 0; i < 64; ++i) {
        int idx = i * 256 + tid;
        int h = idx >> 7;
        float a = in[idx];
        float b = (h + s < Hh) ? in[idx + (s << 7)] : NEGINF;
        out[idx] = fmaxf(a, b);
    }
    __syncthreads();
}

__device__ __forceinline__ void hstep(const float* in, float* out, int s, int tid) {
#pragma unroll 8
    for (int i = 0; i < 64; ++i) {
        int idx = i * 256 + tid;
        int w = idx & (Ww - 1);
        float a = in[idx];
        float b = (w + s < Ww) ? in[idx + s] : NEGINF;
        out[idx] = fmaxf(a, b);
    }
    __syncthreads();
}

// y[h] = max(f26[max(0,h-25)], f26[h])  == -inf-padded centered 51-window max
__device__ __forceinline__ void vcombine(const float* in, float* out, int tid) {
#pragma unroll 8
    for (int i = 0; i < 64; ++i) {
        int idx = i * 256 + tid;
        int h = idx >> 7, w = idx & (Ww - 1);
        int c1 = h - 25; if (c1 < 0) c1 = 0;
        out[idx] = fmaxf(in[(c1 << 7) + w], in[idx]);
    }
    __syncthreads();
}

__device__ __forceinline__ void hcombine(const float* in, float* out, int tid) {
#pragma unroll 8
    for (int i = 0; i < 64; ++i) {
        int idx = i * 256 + tid;
        int h = idx >> 7, w = idx & (Ww - 1);
        int c1 = w - 25; if (c1 < 0) c1 = 0;
        out[idx] = fmaxf(in[(h << 7) + c1], in[idx]);
    }
    __syncthreads();
}

// ---------------------------------------------------------------------------
// Kernel 1: per-plane 51x51 max-pool + local-maxima bitmask + per-plane count.
// One workgroup (256 threads = 8 wave32) per (b,c) plane.  The plane is DMA'd
// into LDS with a single Tensor Data Mover descriptor (TENSORcnt path).
__global__ void __launch_bounds__(256)
nms_plane(const float* __restrict__ x,
          uint32_t* __restrict__ bitmask,   // [NPLANE][512] words
          uint32_t* __restrict__ counts) {  // [NPLANE]
    __shared__ float A[PLANE_PX];
    __shared__ float B[PLANE_PX];
    __shared__ uint32_t wsum[8];

    const int p   = blockIdx.x;
    const int tid = threadIdx.x;
    const float* xp = x + (size_t)p * PLANE_PX;

    // --- TDM: one tensor_load_to_lds moves the whole 64KB plane -------------
    // D# group0 (ISA §8.3): [1:0]=count=1, [63:32]=lds_addr,
    //   [120:64]=global_addr(57b), [127:126]=type=2.
    // D# group1 (ISA §8.4): data_size=4B, tensor_dim0=tensor_dim1=128,
    //   tile_dim0=tile_dim1=128, tensor_dim0_stride=128, dim1_stride=16384.
    if (tid == 0) {
        uint64_t ga = (uint64_t)(uintptr_t)xp;
        u32x4 g0;
        g0.x = 1u;                                   // count=1, user mode
        g0.y = lds_off(&A[0]);                       // lds_addr
        g0.z = (uint32_t)ga;                         // global_addr[31:0]
        g0.w = ((uint32_t)(ga >> 32) & 0x01FFFFFFu)  // global_addr[56:32]
               | (2u << 30);                         // type=2 ("image")
        u32x8 g1;
        g1.s0 = 2u << 16;                            // data_size=2 (4 bytes)
        g1.s1 = (uint32_t)Ww << 16;                  // tensor_dim0 lo16
        g1.s2 = (uint32_t)Hh << 16;                  // dim0 hi=0 | tensor_dim1 lo16
        g1.s3 = (uint32_t)Ww << 16;                  // dim1 hi=0 | tile_dim0
        g1.s4 = (uint32_t)Hh;                        // tile_dim1 | tile_dim2=0
        g1.s5 = (uint32_t)Ww;                        // tensor_dim0_stride lo32
        g1.s6 = (uint32_t)(PLANE_PX & 0xFFFF) << 16; // stride0 hi | dim1_stride lo16
        g1.s7 = (uint32_t)(PLANE_PX >> 16);          // dim1_stride hi
        asm volatile("tensor_load_to_lds %0, %1" :: "s"(g0), "s"(g1) : "memory");
        asm volatile("s_wait_tensorcnt 0x0" ::: "memory");
    }
    __syncthreads();

    // --- vertical 51-window max: A(x)->B(f2)->A(f4)->B(f8)->A(f16)->B(f26)->A(V)
    vstep(A, B, 1, tid);
    vstep(B, A, 2, tid);
    vstep(A, B, 4, tid);
    vstep(B, A, 8, tid);
    vstep(A, B, 10, tid);   // f26 = max(f16[i], f16[i+10])
    vcombine(B, A, tid);

    // --- horizontal 51-window max on V: ends with y in A
    hstep(A, B, 1, tid);
    hstep(B, A, 2, tid);
    hstep(A, B, 4, tid);
    hstep(B, A, 8, tid);
    hstep(A, B, 10, tid);
    hcombine(B, A, tid);

    // --- keep = (x == y) && (x > 0.5); pack via wave32 ballot ---------------
    // x re-read from global: whole tensor (84MB) is L2-resident (192MB L2).
    const int lane = tid & 31;
    uint32_t cnt = 0;
#pragma unroll 8
    for (int i = 0; i < 64; ++i) {
        int idx = i * 256 + tid;                 // lanes cover 32 consecutive px
        float xv = xp[idx];
        bool keep = (xv == A[idx]) && (xv > 0.5f);
        uint32_t m = __builtin_amdgcn_ballot_w32(keep);
        if (lane == 0) {
            bitmask[(size_t)p * 512 + (idx >> 5)] = m;
            cnt += __popc(m);
        }
    }
    if (lane == 0) wsum[tid >> 5] = cnt;
    __syncthreads();
    if (tid == 0) {
        uint32_t t = 0;
#pragma unroll
        for (int w = 0; w < 8; ++w) t += wsum[w];
        counts[p] = t;
    }
}

// ---------------------------------------------------------------------------
// Kernel 2: exclusive scan of 1280 plane counts (256 threads x 5 each).
__global__ void __launch_bounds__(256)
scan_counts(const uint32_t* __restrict__ counts, uint32_t* __restrict__ offsets) {
    __shared__ uint32_t tsum[256];
    const int tid = threadIdx.x;
    uint32_t local[5], s = 0;
#pragma unroll
    for (int i = 0; i < 5; ++i) { local[i] = counts[tid * 5 + i]; s += local[i]; }
    tsum[tid] = s;
    __syncthreads();
    for (int off = 1; off < 256; off <<= 1) {   // inclusive Hillis-Steele
        uint32_t v = tsum[tid];
        uint32_t u = (tid >= off) ? tsum[tid - off] : 0u;
        __syncthreads();
        tsum[tid] = v + u;
        __syncthreads();
    }
    uint32_t run = (tid > 0) ? tsum[tid - 1] : 0u;  // exclusive base
#pragma unroll
    for (int i = 0; i < 5; ++i) { offsets[tid * 5 + i] = run; run += local[i]; }
}

// ---------------------------------------------------------------------------
// Kernel 3: emit (b,c,h,w) rows in exact row-major order from the bitmasks.
// Bitmask pulled into LDS over the ASYNCcnt DMA path; parallel popcount scan.
__global__ void __launch_bounds__(256)
emit_rows(const uint32_t* __restrict__ bitmask,
          const uint32_t* __restrict__ offsets,
          int* __restrict__ out) {
    __shared__ uint32_t msk[512];
    __shared__ uint32_t tsum[256];
    const int p   = blockIdx.x;
    const int tid = threadIdx.x;
    const uint32_t* bm = bitmask + (size_t)p * 512;

    // async global->LDS copy of the 2KB plane bitmask (one b64 per thread)
    {
        uint32_t boff  = (uint32_t)tid * 8u;
        uint32_t laddr = lds_off(&msk[0]) + boff;
        asm volatile("global_load_async_to_lds_b64 %0, %1, %2"
                     :: "v"(laddr), "v"(boff), "s"(bm) : "memory");
        asm volatile("s_wait_asynccnt 0x0" ::: "memory");
    }
    __syncthreads();

    // parallel exclusive scan of word popcounts (2 words per thread)
    uint32_t w0 = msk[2 * tid], w1 = msk[2 * tid + 1];
    uint32_t c0 = __popc(w0), c1 = __popc(w1);
    tsum[tid] = c0 + c1;
    __syncthreads();
    for (int off = 1; off < 256; off <<= 1) {   // inclusive Hillis-Steele
        uint32_t v = tsum[tid];
        uint32_t u = (tid >= off) ? tsum[tid - off] : 0u;
        __syncthreads();
        tsum[tid] = v + u;
        __syncthreads();
    }
    const uint32_t base = offsets[p] + ((tid > 0) ? tsum[tid - 1] : 0u);
    const int bi = p / CHAN, ci = p % CHAN;

    // emit rows for this thread's two words, in row-major rank order
    uint32_t pos = base;
#pragma unroll
    for (int half = 0; half < 2; ++half) {
        uint32_t m = (half == 0) ? w0 : w1;
        int idx0 = (2 * tid + half) << 5;
        while (m) {
            int bit = __builtin_ctz(m);
            m &= m - 1;
            if (pos < MAX_DET) {
                int idx = idx0 + bit;
                ((int4*)out)[pos] = make_int4(bi, ci, idx >> 7, idx & (Ww - 1));
            }
            ++pos;
        }
    }
}

// ---------------------------------------------------------------------------
extern "C" void kernel_launch(void* const* d_in, const int* in_sizes, int n_in,
                              void* d_out, int out_size, void* d_ws, size_t ws_size,
                              hipStream_t stream) {
    (void)in_sizes; (void)n_in; (void)ws_size; (void)out_size;
    const float* x = (const float*)d_in[0];
    int* out = (int*)d_out;                      // MAX_DET x 4 int32

    // workspace layout: counts[1280] | offsets[1280] | pad | bitmask[1280*512]
    uint32_t* counts  = (uint32_t*)d_ws;
    uint32_t* offsets = counts + NPLANE;
    uint32_t* bitmask = (uint32_t*)d_ws + 4096;  // byte offset 16384, ~2.62MB used

    fill_neg1<<<(MAX_DET * 4 + 255) / 256, 256, 0, stream>>>(out, MAX_DET * 4);
    nms_plane<<<NPLANE, 256, 0, stream>>>(x, bitmask, counts);
    scan_counts<<<1, 256, 0, stream>>>(counts, offsets);
    emit_rows<<<NPLANE, 256, 0, stream>>>(bitmask, offsets, out);
}